// Net_3152505995417
// MI455X (gfx1250) — compile-verified
//
#include <hip/hip_runtime.h>

// ---------------------------------------------------------------------------
// Fused brain-connectome GNN forward for MI455X (gfx1250, wave32, WMMA).
// Single workgroup, 1024 threads = 32 waves; each wave owns one 16x16 output
// tile of every GEMM via V_WMMA_F32_16X16X4_F32 (exact f32 math, matching the
// f32 reference). Dense node operator A (116x116) lives in LDS; activations
// live in d_ws (L2-resident). The E x E edge operator is algebraically
// collapsed to a node scatter (never materialized).
//
// Branch-free fragment loads: A rows are clamped (garbage rows only affect
// discarded C rows), K-remainder handled by one masked step (encoder K=122).
// ---------------------------------------------------------------------------

typedef float v2f __attribute__((ext_vector_type(2)));
typedef float v8f __attribute__((ext_vector_type(8)));

constexpr int NN    = 116;                 // nodes
constexpr int HIDF  = 64;                  // hidden
constexpr int ENCF  = 122;                 // raw encoding features
constexpr int EDIMF = 5;                   // edge feature dim
constexpr int NOUT  = 4;                   // classes
constexpr int NTHR  = 1024;                // 32 waves
constexpr float EPSF = 1e-10f;

__device__ __forceinline__ int edge_id(int i, int j) {   // i < j, triu order
  return i * (2 * NN - i - 1) / 2 + (j - i - 1);
}

__device__ __forceinline__ v8f wmma4(v2f a, v2f b, v8f c) {
  // V_WMMA_F32_16X16X4_F32 : D = A(16x4,f32) * B(4x16,f32) + C(16x16,f32)
  return __builtin_amdgcn_wmma_f32_16x16x4_f32(
      /*neg_a=*/false, a, /*neg_b=*/false, b,
      /*c_mod=*/(short)0, c, /*reuse_a=*/false, /*reuse_b=*/false);
}

// One wave computes the C[16x16] tile (rows m0.., column n) of
// A[arows x K](lda) * B[K x ldb](ldb). Branch-free inner loop.
__device__ __forceinline__ v8f gemm_tile(const float* A, int lda, int arows,
                                         const float* B, int ldb,
                                         int K, int m0, int n, int lane) {
  const int r  = lane & 15;
  const int kb = (lane >> 4) * 2;          // lanes 16-31 hold K+2, K+3
  int row = m0 + r;
  if (row >= arows) row = arows - 1;       // clamp only; C rows >= arows are
                                           // discarded at store time
  const float* Arow = A + row * lda;
  v8f acc = {0.f, 0.f, 0.f, 0.f, 0.f, 0.f, 0.f, 0.f};
  const int Kmain = K & ~3;
#pragma unroll
  for (int k0 = 0; k0 < Kmain; k0 += 4) {
    v2f a = *(const v2f*)(Arow + k0 + kb); // 8B-aligned pair load
    v2f b;
    b.x = B[(k0 + kb) * ldb + n];
    b.y = B[(k0 + kb + 1) * ldb + n];
    acc = wmma4(a, b, acc);
  }
  if (Kmain < K) {                         // only the encoder GEMM (K=122)
    const int ka0 = Kmain + kb, ka1 = ka0 + 1;
    v2f a, b;
    a.x = Arow[ka0 < K ? ka0 : K - 1];     // clamped unconditional loads
    a.y = Arow[ka1 < K ? ka1 : K - 1];
    const float bx = B[(ka0 < K ? ka0 : K - 1) * ldb + n];
    const float by = B[(ka1 < K ? ka1 : K - 1) * ldb + n];
    b.x = ka0 < K ? bx : 0.f;              // cndmask, zero B kills the term
    b.y = ka1 < K ? by : 0.f;
    acc = wmma4(a, b, acc);
  }
  return acc;
}

// C VGPR v maps to row m0 + v + 8*(lane/16), col n.
__device__ __forceinline__ void store_tile(float* D, int ldd, int drows,
                                           int m0, int n, int lane, v8f c,
                                           const float* bias, bool do_relu) {
  const int mh = (lane >> 4) * 8;
  const float bv = bias ? bias[n] : 0.f;
  if (m0 + 16 <= drows) {                  // wave-uniform: 7 of 8 M-tiles
#pragma unroll
    for (int v = 0; v < 8; ++v) {
      float val = c[v] + bv;
      if (do_relu) val = fmaxf(val, 0.f);
      D[(m0 + v + mh) * ldd + n] = val;
    }
  } else {                                 // last tile: per-row guard
#pragma unroll
    for (int v = 0; v < 8; ++v) {
      const int m = m0 + v + mh;
      if (m < drows) {
        float val = c[v] + bv;
        if (do_relu) val = fmaxf(val, 0.f);
        D[m * ldd + n] = val;
      }
    }
  }
}

__global__ __launch_bounds__(NTHR)
void brainnet_fused(const float* __restrict__ enc,   // [116,122]
                    const float* __restrict__ ea,    // [6670,5]
                    const float* __restrict__ W_enc, const float* __restrict__ b_enc,
                    const float* __restrict__ W1,    const float* __restrict__ b1,
                    const float* __restrict__ p1,    // [1,5]
                    const float* __restrict__ We,    const float* __restrict__ be,
                    const float* __restrict__ pe,    // [1,64]
                    const float* __restrict__ W2,    const float* __restrict__ b2,
                    const float* __restrict__ p2,    // [1,5]
                    const float* __restrict__ Wl,    const float* __restrict__ bl,
                    float* __restrict__ out,         // [4]
                    float* __restrict__ ws) {
  __shared__ float Amat[NN * NN];        // dense node operator (53.8 KB)
  __shared__ float dvS[NN];              // per-node scalar (x1 @ pe)
  __shared__ float Ssum[NN * EDIMF];     // per-node scatter of normalized Gh
  __shared__ float pooledS[HIDF];

  const int tid  = threadIdx.x;
  const int lane = tid & 31;
  const int wave = tid >> 5;             // 0..31
  const int tm   = wave >> 2;            // 8 M-tiles (M padded to 128)
  const int tn   = wave & 3;             // 4 N-tiles (N = 64)
  const int m0   = tm * 16;
  const int n    = tn * 16 + (lane & 15);

  float* xA = ws;                        // [116,64] x0, later x2
  float* xB = ws + NN * HIDF;            // [116,64] xw / xw2
  float* xC = ws + 2 * NN * HIDF;        // [116,64] x1

  // ---- Phase 1: x0 = enc @ W_enc + b_enc  (K=122: 30 steps + 1 masked) --
  v8f acc = gemm_tile(enc, ENCF, NN, W_enc, HIDF, ENCF, m0, n, lane);
  store_tile(xA, HIDF, NN, m0, n, lane, acc, b_enc, false);

  // ---- Phase 2 (independent): A1[i][j] = edge_attr[e(i,j)] . p1 ---------
  for (int t = tid; t < NN * NN; t += NTHR) {
    const int i = t / NN, j = t % NN;
    if (i < j) {
      const int e = edge_id(i, j);
      float d = 0.f;
#pragma unroll
      for (int cc = 0; cc < EDIMF; ++cc) d += ea[e * EDIMF + cc] * p1[cc];
      Amat[i * NN + j] = d;
      Amat[j * NN + i] = d;
    } else if (i == j) {
      Amat[t] = 0.f;                     // masked diagonal
    }
  }
  __threadfence();
  __syncthreads();

  // ---- Phase 3: xw = x0 @ W1  (K=64, 16 steps) --------------------------
  acc = gemm_tile(xA, HIDF, NN, W1, HIDF, HIDF, m0, n, lane);
  store_tile(xB, HIDF, NN, m0, n, lane, acc, nullptr, false);
  __threadfence();
  __syncthreads();

  // ---- Phase 4: x1 = relu(A1 @ xw + b1)  (K=116, 29 steps, A in LDS) ----
  acc = gemm_tile(Amat, NN, NN, xB, HIDF, NN, m0, n, lane);
  store_tile(xC, HIDF, NN, m0, n, lane, acc, b1, true);
  __threadfence();
  __syncthreads();

  // ---- Phase 5: dv[n] = x1[n,:] . pe ------------------------------------
  if (tid < NN) {
    float s = 0.f;
#pragma unroll 8
    for (int h = 0; h < HIDF; ++h) s += xC[tid * HIDF + h] * pe[h];
    dvS[tid] = s;
  }
  __syncthreads();

  // ---- Phase 6: S[n][c] = sum over edges f incident to n of
  //      Gh[f][c] = (relu(ea[f]) @ We)[c] / (max(0, dv_k, dv_l) + eps)
  //      (deterministic: one thread per (node, channel)) ------------------
  for (int t = tid; t < NN * EDIMF; t += NTHR) {
    const int nn = t / EDIMF, cc = t % EDIMF;
    const float dn = dvS[nn];
    float s = 0.f;
    for (int j = 0; j < NN; ++j) {
      if (j == nn) continue;
      const int i2 = nn < j ? nn : j;
      const int j2 = nn < j ? j : nn;
      const int e  = edge_id(i2, j2);
      float g = 0.f;
#pragma unroll
      for (int k = 0; k < EDIMF; ++k)
        g += fmaxf(ea[e * EDIMF + k], 0.f) * We[k * EDIMF + cc];
      const float cm = fmaxf(0.f, fmaxf(dn, dvS[j]));
      s += g / (cm + EPSF);
    }
    Ssum[t] = s;
  }
  __syncthreads();

  // ---- Phase 7: per edge e=(i,j):
  //      e2 = relu(dv_i*(S_i - Gh_e) + dv_j*(S_j - Gh_e) + be)
  //      A2[i][j] = e2 . p2   (edge features never materialized) ----------
  for (int t = tid; t < NN * NN; t += NTHR) {
    const int i = t / NN, j = t % NN;
    if (i < j) {
      const int e = edge_id(i, j);
      const float di = dvS[i], dj = dvS[j];
      const float inv = 1.f / (fmaxf(0.f, fmaxf(di, dj)) + EPSF);
      float d2 = 0.f;
#pragma unroll
      for (int cc = 0; cc < EDIMF; ++cc) {
        float g = 0.f;
#pragma unroll
        for (int k = 0; k < EDIMF; ++k)
          g += fmaxf(ea[e * EDIMF + k], 0.f) * We[k * EDIMF + cc];
        const float gh = g * inv;
        const float v = di * (Ssum[i * EDIMF + cc] - gh) +
                        dj * (Ssum[j * EDIMF + cc] - gh) + be[cc];
        d2 += fmaxf(v, 0.f) * p2[cc];
      }
      Amat[i * NN + j] = d2;
      Amat[j * NN + i] = d2;             // diagonal stays 0 from Phase 2
    }
  }
  __syncthreads();

  // ---- Phase 8: xw2 = x1 @ W2  (K=64) -----------------------------------
  acc = gemm_tile(xC, HIDF, NN, W2, HIDF, HIDF, m0, n, lane);
  store_tile(xB, HIDF, NN, m0, n, lane, acc, nullptr, false);
  __threadfence();
  __syncthreads();

  // ---- Phase 9: x2 = A2 @ xw2 + b2  (no relu) ---------------------------
  acc = gemm_tile(Amat, NN, NN, xB, HIDF, NN, m0, n, lane);
  store_tile(xA, HIDF, NN, m0, n, lane, acc, b2, false);
  __threadfence();
  __syncthreads();

  // ---- Phase 10: mean-pool over nodes, classifier head ------------------
  if (tid < HIDF) {
    float s = 0.f;
#pragma unroll 8
    for (int a = 0; a < NN; ++a) s += xA[a * HIDF + tid];
    pooledS[tid] = s * (1.0f / NN);
  }
  __syncthreads();
  if (tid < NOUT) {
    float s = bl[tid];
#pragma unroll 8
    for (int h = 0; h < HIDF; ++h) s += pooledS[h] * Wl[h * NOUT + tid];
    out[tid] = s;
  }
}

extern "C" void kernel_launch(void* const* d_in, const int* in_sizes, int n_in,
                              void* d_out, int out_size, void* d_ws, size_t ws_size,
                              hipStream_t stream) {
  // setup_inputs() order:
  // 0 encoding_raw, 1 edge_attr, 2 edge_index (unused; indices computed
  // analytically), 3 W_enc, 4 b_enc, 5 W1, 6 b1, 7 p1, 8 We, 9 be, 10 pe,
  // 11 W2, 12 b2, 13 p2, 14 Wl, 15 bl
  const float* enc   = (const float*)d_in[0];
  const float* ea    = (const float*)d_in[1];
  const float* W_enc = (const float*)d_in[3];
  const float* b_enc = (const float*)d_in[4];
  const float* W1    = (const float*)d_in[5];
  const float* b1    = (const float*)d_in[6];
  const float* p1    = (const float*)d_in[7];
  const float* We    = (const float*)d_in[8];
  const float* be    = (const float*)d_in[9];
  const float* pe    = (const float*)d_in[10];
  const float* W2    = (const float*)d_in[11];
  const float* b2    = (const float*)d_in[12];
  const float* p2    = (const float*)d_in[13];
  const float* Wl    = (const float*)d_in[14];
  const float* bl    = (const float*)d_in[15];

  brainnet_fused<<<1, NTHR, 0, stream>>>(
      enc, ea, W_enc, b_enc, W1, b1, p1, We, be, pe, W2, b2, p2, Wl, bl,
      (float*)d_out, (float*)d_ws);
}